// GeneralConvNet_64132451664027
// MI455X (gfx1250) — compile-verified
//
#include <hip/hip_runtime.h>

// Problem constants (from reference)
#define NN 100000
#define EE 1200000
#define CIN 128
#define COUT 64
#define EDIM 16

typedef __attribute__((ext_vector_type(2))) float v2f;
typedef __attribute__((ext_vector_type(8))) float v8f;

// ---------------------------------------------------------------------------
// fill zeros
__global__ void fill_zero_kernel(float* __restrict__ p, int n) {
    int i = blockIdx.x * blockDim.x + threadIdx.x;
    if (i < n) p[i] = 0.0f;
}

// ---------------------------------------------------------------------------
// degree: deg[dst] += 1   (once per launch; edge topology is layer-invariant)
__global__ void edge_deg_kernel(const int* __restrict__ ei, float* __restrict__ deg) {
    int e = blockIdx.x * blockDim.x + threadIdx.x;
    if (e >= EE) return;
    int dst = ei[EE + e];
    unsafeAtomicAdd(&deg[dst], 1.0f);
}

// eagg[dst, k] += edge_attr[e, k]   (once per launch)
__global__ void edge_eagg_kernel(const int* __restrict__ ei,
                                 const float* __restrict__ eattr,
                                 float* __restrict__ eagg) {
    int idx = blockIdx.x * blockDim.x + threadIdx.x;
    if (idx >= EE * EDIM) return;
    int e = idx >> 4;
    int k = idx & 15;
    int dst = ei[EE + e];
    unsafeAtomicAdd(&eagg[dst * EDIM + k], eattr[idx]);
}

// ---------------------------------------------------------------------------
// C[M x 64] = A[M x K] @ W[K x 64], fp32 WMMA 16x16x4.
// One wave per 16x16 C tile; block = 128 thr = 4 waves = one 16-row strip.
// A frag (16x4 f32): lane<16 holds A[m][k+0..1] in v0/v1; lane>=16 holds K+2..3.
// B frag (4x16):     v0: rows K/K+2 striped over lanes; v1: rows K+1/K+3.
// C/D layout: VGPR r -> row r (lanes 0-15) and row r+8 (lanes 16-31).
__global__ __launch_bounds__(128)
void gemm_wmma_f32(const float* __restrict__ A, const float* __restrict__ W,
                   float* __restrict__ C, int K) {
    const int tid  = threadIdx.x;
    const int lane = tid & 31;
    const int wave = tid >> 5;
    const int row0 = blockIdx.x << 4;   // 16-row strip
    const int n0   = wave << 4;         // column tile: 0,16,32,48
    const int m    = lane & 15;
    const int kb   = (lane >> 4) << 1;  // 0 or 2

    const float* Arow = A + (size_t)(row0 + m) * K;
    v8f acc = {0.f, 0.f, 0.f, 0.f, 0.f, 0.f, 0.f, 0.f};

    for (int k = 0; k < K; k += 4) {
        v2f a = *(const v2f*)(Arow + k + kb);                 // 8B load per lane
        v2f b;
        b.x = W[(size_t)(k + kb)     * COUT + n0 + m];
        b.y = W[(size_t)(k + kb + 1) * COUT + n0 + m];
        acc = __builtin_amdgcn_wmma_f32_16x16x4_f32(
            false, a, false, b, (short)0, acc, false, false);
    }

    const int rbase = row0 + ((lane >> 4) << 3);
    float* Cp = C + (size_t)rbase * COUT + n0 + m;
#pragma unroll
    for (int r = 0; r < 8; ++r) Cp[(size_t)r * COUT] = acc[r];
}

// ---------------------------------------------------------------------------
// agg[dst, :] += xm[src, :]   -- 16 threads per edge, float4 each.
// agg is 25.6 MB -> resident in 192 MB L2, fp32 atomics resolve in L2.
__global__ __launch_bounds__(256)
void scatter_add_kernel(const int* __restrict__ ei,
                        const float* __restrict__ xm,
                        float* __restrict__ agg) {
    int idx = blockIdx.x * blockDim.x + threadIdx.x;
    if (idx >= EE * 16) return;
    int e = idx >> 4;
    int q = (idx & 15) << 2;            // channel group base: 0,4,...,60
    int src = ei[e];
    int dst = ei[EE + e];
    const float4 v = *(const float4*)(xm + (size_t)src * COUT + q);
    float* a = agg + (size_t)dst * COUT + q;
    unsafeAtomicAdd(a + 0, v.x);
    unsafeAtomicAdd(a + 1, v.y);
    unsafeAtomicAdd(a + 2, v.z);
    unsafeAtomicAdd(a + 3, v.w);
}

// ---------------------------------------------------------------------------
// h[n,c] = ELU( agg[n,c] + self[n,c] + eagg[n,:]@We[:,c] + deg[n]*(bm[c]+be[c]) + bs[c] )
__global__ __launch_bounds__(256)
void epilogue_kernel(const float* __restrict__ agg, const float* __restrict__ sf,
                     const float* __restrict__ eagg, const float* __restrict__ deg,
                     const float* __restrict__ We, const float* __restrict__ bm,
                     const float* __restrict__ be, const float* __restrict__ bs,
                     float* __restrict__ out) {
    int idx = blockIdx.x * blockDim.x + threadIdx.x;
    if (idx >= NN * COUT) return;
    int n = idx >> 6;
    int c = idx & 63;
    const float* ea = eagg + n * EDIM;
    float dot = 0.f;
#pragma unroll
    for (int k = 0; k < EDIM; ++k) dot = fmaf(ea[k], We[k * COUT + c], dot);
    float v = agg[idx] + sf[idx] + dot + deg[n] * (bm[c] + be[c]) + bs[c];
    out[idx] = (v > 0.f) ? v : expm1f(v);
}

// ---------------------------------------------------------------------------
// Final head: logit = h3[n,:]@Wl + bl; write sigmoid, feat copy, concat(feat,logit).
// One wave (32 lanes) per node; 8 nodes per 256-thread block. N = 12500*8 exactly.
__global__ __launch_bounds__(256)
void final_kernel(const float* __restrict__ feat, const float* __restrict__ h3,
                  const float* __restrict__ Wl, const float* __restrict__ bl,
                  float* __restrict__ out) {
    int node = blockIdx.x * 8 + (threadIdx.x >> 5);
    int lane = threadIdx.x & 31;
    const float* f = feat + (size_t)node * COUT;
    const float* h = h3 + (size_t)node * COUT;
    float f0 = f[lane], f1 = f[lane + 32];
    float h0 = h[lane], h1 = h[lane + 32];
    float part = fmaf(h0, Wl[lane], h1 * Wl[lane + 32]);
#pragma unroll
    for (int off = 16; off >= 1; off >>= 1)
        part += __shfl_xor(part, off, 32);   // wave32 reduction
    float logit = part + bl[0];

    float* r1 = out + NN + (size_t)node * COUT;               // feat [N,64]
    r1[lane] = f0; r1[lane + 32] = f1;
    float* r2 = out + NN + (size_t)NN * COUT + (size_t)node * (COUT + 1); // concat [N,65]
    r2[lane] = f0; r2[lane + 32] = f1;
    if (lane == 0) {
        out[node] = 1.f / (1.f + __expf(-logit) * 0.f + expf(-logit) * 1.f - 0.f) ; // sigmoid
        out[node] = 1.f / (1.f + expf(-logit));
        r2[COUT] = logit;
    }
}

// ---------------------------------------------------------------------------
extern "C" void kernel_launch(void* const* d_in, const int* in_sizes, int n_in,
                              void* d_out, int out_size, void* d_ws, size_t ws_size,
                              hipStream_t stream) {
    const float* x      = (const float*)d_in[0];
    const int*   ei     = (const int*)d_in[1];     // [2,E] row-major: src row then dst row
    const float* eattr  = (const float*)d_in[2];
    // per-layer params start at 3, six tensors per layer: Wm,bm,We,be,Ws,bs
    const float* Wm[3], *bm[3], *We[3], *be[3], *Ws[3], *bs[3];
    for (int l = 0; l < 3; ++l) {
        int b = 3 + l * 6;
        Wm[l] = (const float*)d_in[b + 0];
        bm[l] = (const float*)d_in[b + 1];
        We[l] = (const float*)d_in[b + 2];
        be[l] = (const float*)d_in[b + 3];
        Ws[l] = (const float*)d_in[b + 4];
        bs[l] = (const float*)d_in[b + 5];
    }
    const float* Wl = (const float*)d_in[21];
    const float* bl = (const float*)d_in[22];
    float* out = (float*)d_out;

    // workspace carve-up (floats)
    float* ws   = (float*)d_ws;
    float* xm   = ws;                         // N*64  (reused for self-term after scatter)
    float* agg  = xm   + (size_t)NN * COUT;   // N*64
    float* h1   = agg  + (size_t)NN * COUT;   // N*64
    float* feat = h1   + (size_t)NN * COUT;   // N*64
    float* h3   = feat + (size_t)NN * COUT;   // N*64
    float* eagg = h3   + (size_t)NN * COUT;   // N*16
    float* deg  = eagg + (size_t)NN * EDIM;   // N      (eagg+deg contiguous)

    const int T = 256;
    // ---- once-per-launch graph reductions ----
    fill_zero_kernel<<<(NN * (EDIM + 1) + T - 1) / T, T, 0, stream>>>(eagg, NN * (EDIM + 1));
    edge_deg_kernel<<<(EE + T - 1) / T, T, 0, stream>>>(ei, deg);
    edge_eagg_kernel<<<(EE * EDIM + T - 1) / T, T, 0, stream>>>(ei, eattr, eagg);

    const float* layer_in[3]  = { x, h1, feat };
    const int    layer_K[3]   = { CIN, COUT, COUT };
    float*       layer_out[3] = { h1, feat, h3 };

    for (int l = 0; l < 3; ++l) {
        fill_zero_kernel<<<(NN * COUT + T - 1) / T, T, 0, stream>>>(agg, NN * COUT);
        // xm = x_l @ Wm  (WMMA, fp32)
        gemm_wmma_f32<<<NN / 16, 128, 0, stream>>>(layer_in[l], Wm[l], xm, layer_K[l]);
        // agg[dst] += xm[src]
        scatter_add_kernel<<<(EE * 16 + T - 1) / T, T, 0, stream>>>(ei, xm, agg);
        // self term reuses xm buffer (stream order serializes with scatter)
        gemm_wmma_f32<<<NN / 16, 128, 0, stream>>>(layer_in[l], Ws[l], xm, layer_K[l]);
        // combine + ELU
        epilogue_kernel<<<(NN * COUT + T - 1) / T, T, 0, stream>>>(
            agg, xm, eagg, deg, We[l], bm[l], be[l], bs[l], layer_out[l]);
    }

    final_kernel<<<NN / 8, 256, 0, stream>>>(feat, h3, Wl, bl, out);
}